// MSDAv1_46394236731949
// MI455X (gfx1250) — compile-verified
//
#include <hip/hip_runtime.h>

// ---------------------------------------------------------------------------
// MSDA (multi-scale deformable attention) for gfx1250 / MI455X, fp32 path.
// GEMMs run on V_WMMA_F32_16X16X4_F32 (fp32 in / fp32 accumulate, faithful to
// the fp32 reference). Tiles are staged into LDS with
// GLOBAL_LOAD_ASYNC_TO_LDS_B128 (ASYNCcnt DMA) when the toolchain exposes the
// builtin; otherwise via float4. Gather stage is wave32 VALU with coalesced
// 128B per-corner loads (lane == channel d, D == 32 == wave size).
// ---------------------------------------------------------------------------

typedef __attribute__((ext_vector_type(2))) float v2f;
typedef __attribute__((ext_vector_type(8))) float v8f;

#define Bc   4
#define Qc   2048
#define Lc   4
#define Hc   8
#define Pc   4
#define Dc   32
#define Sc   21760
#define Kc   256        // inner dim of every GEMM

#define TM   128        // workgroup tile rows
#define TN   128        // workgroup tile cols
#define TK   32         // K chunk staged in LDS
#define LDA  (TK + 4)   // +4 floats: keeps 16B alignment for B128 async DMA
#define LDB  (TN + 4)   //           and 16 distinct banks for fragment reads

#if defined(__gfx1250__) && __has_builtin(__builtin_amdgcn_global_load_async_to_lds_b128)
#define USE_ASYNC_LDS 1
// Builtin expects int4 ("vector_size(16) int") pointers, global src / LDS dst.
typedef int v4i __attribute__((vector_size(16)));
typedef __attribute__((address_space(1))) v4i g_v4i;
typedef __attribute__((address_space(3))) v4i l_v4i;
#endif

__constant__ int kMapW[Lc]     = {128, 64, 32, 16};
__constant__ int kMapH[Lc]     = {128, 64, 32, 16};
__constant__ int kMapStart[Lc] = {0, 16384, 20480, 21504};

// ---------------------------------------------------------------------------
// C[M x N] = A[M x 256] * W[256 x N] + bias[N]     (all fp32, row-major)
// M must be a multiple of 128, N a multiple of 128 (true for all call sites).
// 256 threads = 8 wave32; wave grid 4x2, each wave owns a 32x64 C tile
// (2x4 WMMA accumulators) -> 8 WMMAs per K=4 step from 12 LDS dwords.
// ---------------------------------------------------------------------------
__global__ __launch_bounds__(256)
void msda_gemm_wmma(const float* __restrict__ A, const float* __restrict__ Wm,
                    const float* __restrict__ bias, float* __restrict__ C,
                    int N) {
  __shared__ float As[TM * LDA];
  __shared__ float Bs[TK * LDB];

  const int tid  = threadIdx.x;
  const int lane = tid & 31;
  const int wid  = tid >> 5;
  const int half = lane >> 4;     // 0: lanes 0-15, 1: lanes 16-31
  const int r    = lane & 15;

  const int ntn = N / TN;
  const int m0  = (blockIdx.x / ntn) * TM;
  const int n0  = (blockIdx.x % ntn) * TN;
  const int wm  = (wid & 3) * 32;   // wave row offset inside tile
  const int wn  = (wid >> 2) * 64;  // wave col offset inside tile

  v8f acc[2][4] = {};

  for (int k0 = 0; k0 < Kc; k0 += TK) {
    // ---- Stage A tile (TM x TK) and W tile (TK x TN) into LDS -------------
#ifdef USE_ASYNC_LDS
    for (int i = tid; i < TM * (TK / 4); i += 256) {          // 4 per thread
      const int row = i >> 3;
      const int c4  = (i & 7) * 4;
      __builtin_amdgcn_global_load_async_to_lds_b128(
          (g_v4i*)(A + (size_t)(m0 + row) * Kc + k0 + c4),
          (l_v4i*)&As[row * LDA + c4], 0, 0);
    }
    for (int i = tid; i < TK * (TN / 4); i += 256) {          // 4 per thread
      const int row = i >> 5;
      const int c4  = (i & 31) * 4;
      __builtin_amdgcn_global_load_async_to_lds_b128(
          (g_v4i*)(Wm + (size_t)(k0 + row) * N + n0 + c4),
          (l_v4i*)&Bs[row * LDB + c4], 0, 0);
    }
    asm volatile("s_wait_asynccnt 0" ::: "memory");
#else
    for (int i = tid; i < TM * (TK / 4); i += 256) {
      const int row = i >> 3;
      const int c4  = (i & 7) * 4;
      *(float4*)&As[row * LDA + c4] =
          *(const float4*)(A + (size_t)(m0 + row) * Kc + k0 + c4);
    }
    for (int i = tid; i < TK * (TN / 4); i += 256) {
      const int row = i >> 5;
      const int c4  = (i & 31) * 4;
      *(float4*)&Bs[row * LDB + c4] =
          *(const float4*)(Wm + (size_t)(k0 + row) * N + n0 + c4);
    }
#endif
    __syncthreads();

    // ---- 8 K-steps of V_WMMA_F32_16X16X4_F32 ------------------------------
    #pragma unroll
    for (int kk = 0; kk < TK; kk += 4) {
      const int kb = kk + half * 2;   // A/B: lanes>=16 hold K=k+2,k+3
      v2f a0, a1, b[4];
      a0.x = As[(wm + r)      * LDA + kb];
      a0.y = As[(wm + r)      * LDA + kb + 1];
      a1.x = As[(wm + 16 + r) * LDA + kb];
      a1.y = As[(wm + 16 + r) * LDA + kb + 1];
      #pragma unroll
      for (int j = 0; j < 4; ++j) {
        b[j].x = Bs[(kb)     * LDB + wn + j * 16 + r];
        b[j].y = Bs[(kb + 1) * LDB + wn + j * 16 + r];
      }
      #pragma unroll
      for (int j = 0; j < 4; ++j) {
        acc[0][j] = __builtin_amdgcn_wmma_f32_16x16x4_f32(false, a0, false, b[j],
                                                          (short)0, acc[0][j], false, false);
        acc[1][j] = __builtin_amdgcn_wmma_f32_16x16x4_f32(false, a1, false, b[j],
                                                          (short)0, acc[1][j], false, false);
      }
    }
    __syncthreads();
  }

  // Store C + bias. C layout: VGPR v -> M = v + half*8, N = n-base + r.
  #pragma unroll
  for (int i = 0; i < 2; ++i) {
    #pragma unroll
    for (int j = 0; j < 4; ++j) {
      const int col = n0 + wn + j * 16 + r;
      const float bv = bias[col];
      #pragma unroll
      for (int v = 0; v < 8; ++v) {
        const int row = m0 + wm + i * 16 + half * 8 + v;
        C[(size_t)row * N + col] = acc[i][j][v] + bv;
      }
    }
  }
}

// ---------------------------------------------------------------------------
// Fused softmax + bilinear deformable aggregation.
// One wave per (b,q,h); lane = channel d (D == 32 == wave32 width), so each
// corner gather is one coalesced 128B line from the value tensor (B,S,H,D).
// The whole value tensor (89MB) is L2-resident (192MB), so gathers are
// L2-bandwidth bound, not HBM bound.
// ---------------------------------------------------------------------------
__global__ __launch_bounds__(256)
void msda_aggregate(const float* __restrict__ off,     // (B,Q, H*L*P*2)
                    const float* __restrict__ priors,  // (B,Q,L,2)
                    const float* __restrict__ logits,  // (B,Q, H*L*P)
                    const float* __restrict__ value,   // (B,S,H,D)
                    float* __restrict__ wv) {          // (B,Q, H*D)
  const int wave = (blockIdx.x * blockDim.x + threadIdx.x) >> 5;
  const int lane = threadIdx.x & 31;
  const int h  = wave & (Hc - 1);
  const int bq = wave >> 3;           // 0 .. B*Q-1
  const int b  = bq >> 11;            // Q = 2048

  // softmax over L*P = 16 attention logits for this head (lane-redundant,
  // broadcast loads hit L0/L2)
  const float* lg = logits + (size_t)bq * (Hc * Lc * Pc) + h * (Lc * Pc);
  float aw[Lc * Pc];
  float mx = -3.402823466e+38f;
  #pragma unroll
  for (int j = 0; j < Lc * Pc; ++j) { aw[j] = lg[j]; mx = fmaxf(mx, aw[j]); }
  float ssum = 0.f;
  #pragma unroll
  for (int j = 0; j < Lc * Pc; ++j) { aw[j] = __expf(aw[j] - mx); ssum += aw[j]; }
  const float inv = 1.f / ssum;

  const float* offp = off + (size_t)bq * (Hc * Lc * Pc * 2) + h * (Lc * Pc * 2);
  const float* prp  = priors + (size_t)bq * (Lc * 2);

  float acc = 0.f;
  #pragma unroll
  for (int l = 0; l < Lc; ++l) {
    const int   Wi = kMapW[l], Hi = kMapH[l], st = kMapStart[l];
    const float Wf = (float)Wi, Hf = (float)Hi;
    const float px = prp[l * 2 + 0];
    const float py = prp[l * 2 + 1];
    #pragma unroll
    for (int p = 0; p < Pc; ++p) {
      const float ox = offp[(l * Pc + p) * 2 + 0];
      const float oy = offp[(l * Pc + p) * 2 + 1];
      // loc = prior + off / [W,H];  ix = loc_x*W - 0.5, iy = loc_y*H - 0.5
      const float ix = (px + ox / Wf) * Wf - 0.5f;
      const float iy = (py + oy / Hf) * Hf - 0.5f;
      const float x0f = floorf(ix), y0f = floorf(iy);
      const int   x0 = (int)x0f,   y0 = (int)y0f;
      const float fx = ix - x0f,   fy = iy - y0f;
      const float a  = aw[l * Pc + p] * inv;
      const float w00 = (1.f - fx) * (1.f - fy) * a;
      const float w01 = fx * (1.f - fy) * a;
      const float w10 = (1.f - fx) * fy * a;
      const float w11 = fx * fy * a;
      #pragma unroll
      for (int dy = 0; dy < 2; ++dy) {
        #pragma unroll
        for (int dx = 0; dx < 2; ++dx) {
          const int xi = x0 + dx, yi = y0 + dy;
          const float wgt = dy ? (dx ? w11 : w10) : (dx ? w01 : w00);
          if (xi >= 0 && xi < Wi && yi >= 0 && yi < Hi) {
            const int idx = st + yi * Wi + xi;
            const float g = value[(((size_t)(b * Sc + idx)) * Hc + h) * Dc + lane];
            acc += wgt * g;
          }
        }
      }
    }
  }
  wv[(size_t)bq * (Hc * Dc) + h * Dc + lane] = acc;
}

// ---------------------------------------------------------------------------
extern "C" void kernel_launch(void* const* d_in, const int* in_sizes, int n_in,
                              void* d_out, int out_size, void* d_ws, size_t ws_size,
                              hipStream_t stream) {
  (void)in_sizes; (void)n_in; (void)out_size; (void)ws_size;
  const float* in_feats = (const float*)d_in[0];   // (B,Q,256)
  const float* priors   = (const float*)d_in[1];   // (B,Q,L,2)
  const float* samp     = (const float*)d_in[2];   // (B,S,256)
  // d_in[3], d_in[4]: map shapes / start ids -> compile-time constants here
  const float* W_off  = (const float*)d_in[5];
  const float* b_off  = (const float*)d_in[6];
  const float* W_attn = (const float*)d_in[7];
  const float* b_attn = (const float*)d_in[8];
  const float* W_val  = (const float*)d_in[9];
  const float* b_val  = (const float*)d_in[10];
  const float* W_out  = (const float*)d_in[11];
  const float* b_out  = (const float*)d_in[12];
  float* out = (float*)d_out;

  // Workspace carve-up: value | off | attn | wv   (~110 MB total)
  char* ws = (char*)d_ws;
  const size_t sz_value = (size_t)Bc * Sc * 256 * sizeof(float);  // 89,128,960
  const size_t sz_off   = (size_t)Bc * Qc * 256 * sizeof(float);  //  8,388,608
  const size_t sz_attn  = (size_t)Bc * Qc * 128 * sizeof(float);  //  4,194,304
  float* value = (float*)ws;
  float* off   = (float*)(ws + sz_value);
  float* attn  = (float*)(ws + sz_value + sz_off);
  float* wv    = (float*)(ws + sz_value + sz_off + sz_attn);

  const dim3 blk(256);
  const int MQ = Bc * Qc;       // 8192
  const int MV = Bc * Sc;       // 87040

  // 1) value = sample_feats @ W_val + b_val          (87040 x 256)
  msda_gemm_wmma<<<(MV / TM) * (256 / TN), blk, 0, stream>>>(samp, W_val, b_val, value, 256);
  // 2) off = in_feats @ W_off + b_off                (8192 x 256)
  msda_gemm_wmma<<<(MQ / TM) * (256 / TN), blk, 0, stream>>>(in_feats, W_off, b_off, off, 256);
  // 3) attn logits = in_feats @ W_attn + b_attn      (8192 x 128)
  msda_gemm_wmma<<<(MQ / TM) * (128 / TN), blk, 0, stream>>>(in_feats, W_attn, b_attn, attn, 128);
  // 4) softmax + bilinear gather -> wv               (8192 x 256)
  msda_aggregate<<<(MQ * Hc) / 8, blk, 0, stream>>>(off, priors, attn, value, wv);
  // 5) out = wv @ W_out + b_out                      (8192 x 256)
  msda_gemm_wmma<<<(MQ / TM) * (256 / TN), blk, 0, stream>>>(wv, W_out, b_out, out, 256);
}